// TTN_PoolTI_2by2to1_40638980555113
// MI455X (gfx1250) — compile-verified
//
#include <hip/hip_runtime.h>
#include <hip/hip_bf16.h>

typedef float v2f __attribute__((ext_vector_type(2)));
typedef float v8f __attribute__((ext_vector_type(8)));

#define WAVES_PER_BLOCK 8
#define BLOCK (WAVES_PER_BLOCK * 32)

// Shapes: x(32,8,4,128,128) f32, tensors(8,8,4,4,4,4,4) f32, bias(8,4) f32
// out(32,8,4,64,64) f32.
// GEMM view: out[m, pix] = sum_{c,q} T[m, c*256+q] * w[c,q,pix],
//   m = p*4+o (32), pix = n*4096 + X*64 + Y (131072), K = 2048.
__global__ __launch_bounds__(BLOCK) void ttn_pool_wmma_kernel(
    const float* __restrict__ x,
    const float* __restrict__ T,
    const float* __restrict__ bias,
    float* __restrict__ out)
{
    // A-chunk for one input channel c: Alds[pair][m] = (A[m][2*pair], A[m][2*pair+1])
    // pair in [0,128), m in [0,32). 32 KB.
    __shared__ v2f Alds[128 * 32];

    const int tid   = threadIdx.x;
    const int lane  = tid & 31;
    const int wave  = tid >> 5;
    const int laneN = lane & 15;   // pixel column within tile; also A row (M)
    const int h     = lane >> 4;   // K-half: lanes 16-31 carry K=2,3 of each 4-wide step

    // 16-pixel tile for this wave
    const int tile = blockIdx.x * WAVES_PER_BLOCK + wave;
    const int flat = tile * 16 + laneN;        // n*4096 + X*64 + Y  (16 | 64, no straddle)
    const int n  = flat >> 12;
    const int XY = flat & 4095;
    const int X  = XY >> 6;
    const int Y  = XY & 63;

    const int sx0 = 2 * X, sy0 = 2 * Y;
    const float* ximg = x + n * (8 * 4 * 128 * 128);   // x[n][c][dd][sx][sy]

    v8f acc0 = {};   // output rows m = 0..15
    v8f acc1 = {};   // output rows m = 16..31

    for (int c = 0; c < 8; ++c) {
        __syncthreads();
        // Stage T chunk for channel c into LDS in WMMA-A order.
        // A[m][q] = T_flat[ ((c*8 + p)*256 + q)*4 + o ],  p = m>>2, o = m&3.
        for (int e = tid; e < 4096; e += BLOCK) {
            const int pair = e >> 5;
            const int m    = e & 31;
            const int p    = m >> 2;
            const int o    = m & 3;
            const int g    = ((c * 8 + p) * 256 + 2 * pair) * 4 + o;
            v2f val;
            val.x = T[g];
            val.y = T[g + 4];   // q+1 -> +4 floats (o stride is 1, q stride is 4)
            Alds[e] = val;
        }
        __syncthreads();

        // Per-pixel x pieces for this channel:
        //   xa[dd]=x[2X][2Y], xc[dd]=x[2X][2Y+1], xb[dd]=x[2X+1][2Y], xd[dd]=x[2X+1][2Y+1]
        float a[4], b[4], cc[4], dall[4];
        const float* xcb = ximg + c * (4 * 128 * 128);
        #pragma unroll
        for (int dd = 0; dd < 4; ++dd) {
            const float* pr = xcb + dd * (128 * 128) + sx0 * 128 + sy0;
            a[dd]    = pr[0];       // b64-pairable with pr[1]
            cc[dd]   = pr[1];
            b[dd]    = pr[128];
            dall[dd] = pr[129];
        }
        const float d0 = h ? dall[2] : dall[0];   // l = 2h
        const float d1 = h ? dall[3] : dall[1];   // l = 2h+1

        // K loop over q: step s = i*16 + j*4 + k covers q = 4s..4s+3.
        // Lane supplies w[q=4s+2h+{0,1}] = a_i*b_j*c_k * d_{2h,2h+1}.
        int s = 0;
        #pragma unroll
        for (int i = 0; i < 4; ++i) {
            #pragma unroll
            for (int j = 0; j < 4; ++j) {
                const float ab = a[i] * b[j];
                #pragma unroll
                for (int k = 0; k < 4; ++k) {
                    const float abc = ab * cc[k];
                    v2f Bv;
                    Bv.x = abc * d0;
                    Bv.y = abc * d1;
                    const int pairIdx = 2 * s + h;
                    const v2f A0 = Alds[pairIdx * 32 + laneN];        // rows 0..15
                    const v2f A1 = Alds[pairIdx * 32 + laneN + 16];   // rows 16..31
                    acc0 = __builtin_amdgcn_wmma_f32_16x16x4_f32(
                        false, A0, false, Bv, (short)0, acc0, false, false);
                    acc1 = __builtin_amdgcn_wmma_f32_16x16x4_f32(
                        false, A1, false, Bv, (short)0, acc1, false, false);
                    ++s;
                }
            }
        }
    }

    // D layout: lanes 0-15 VGPR r -> (M=r, N=lane); lanes 16-31 -> (M=r+8, N=lane-16).
    // out[(n*32 + m)*4096 + X*64 + Y], bias_flat[m] with m = p*4+o.
    float* outp = out + (n * 32) * 4096 + X * 64 + Y;
    const int mb0 = h * 8;        // acc0 rows m = mb0..mb0+7
    const int mb1 = 16 + h * 8;   // acc1 rows
    #pragma unroll
    for (int r = 0; r < 8; ++r) {
        const int m0 = mb0 + r;
        const int m1 = mb1 + r;
        outp[m0 * 4096] = acc0[r] + bias[m0];
        outp[m1 * 4096] = acc1[r] + bias[m1];
    }
}

extern "C" void kernel_launch(void* const* d_in, const int* in_sizes, int n_in,
                              void* d_out, int out_size, void* d_ws, size_t ws_size,
                              hipStream_t stream) {
    const float* x    = (const float*)d_in[0];
    const float* T    = (const float*)d_in[1];
    const float* bias = (const float*)d_in[2];
    float* out        = (float*)d_out;

    // 131072 pixels / (8 waves * 16 pixels) = 1024 blocks
    ttn_pool_wmma_kernel<<<1024, BLOCK, 0, stream>>>(x, T, bias, out);
}